// AdaptiveMultiWIRE_81741817577968
// MI455X (gfx1250) — compile-verified
//
#include <hip/hip_runtime.h>

typedef __attribute__((ext_vector_type(2))) float v2f;
typedef __attribute__((ext_vector_type(8))) float v8f;

#define NCHAN   8
#define NPTS    32768
#define HID     128
#define OUTF    3
#define OMEGA0  30.0f
#define S0SQ    100.0f

#define TILE_M  64
#define KT      16
#define NKT     (HID / KT)          /* 8 k-tiles */
#define ZSTR    130                 /* padded LDS stride (keeps v2f-even) */
#define PLANE   (KT * HID)          /* 2048 floats per plane */

/* ---- LDS partition (float offsets) ---- */
#define OFF_ZRE   0
#define OFF_ZIM   (OFF_ZRE + TILE_M * ZSTR)          /* 8320  */
#define OFF_BT    (OFF_ZIM + TILE_M * ZSTR)          /* 16640 */
#define BT_FLOATS (2 * 4 * PLANE)                    /* 16384: 2 buf x 4 planes */
#define OFF_MAG   OFF_BT                             /* reused after GEMM */
#define OFF_WFR   (OFF_BT + BT_FLOATS)               /* 33024 */
#define OFF_WFI   (OFF_WFR + HID * OUTF)             /* 33408 */
#define OFF_W0    (OFF_WFI + HID * OUTF)             /* 33792 */
#define OFF_XLS   (OFF_W0 + 4 * HID)                 /* 34304 */
#define OFF_BIDX  (OFF_XLS + 2 * TILE_M)             /* 34432 */
#define SMEM_FLOATS (OFF_BIDX + TILE_M)              /* 34496 */
#define SMEM_BYTES  (SMEM_FLOATS * 4)                /* 137984 B: 2 blocks/WGP */

__global__ __launch_bounds__(256)
void wire_fused_kernel(const float* __restrict__ inp,
                       const int*   __restrict__ indices,
                       const int*   __restrict__ model_idx,
                       const int*   __restrict__ bias_idx,
                       const float* __restrict__ W0a, const float* __restrict__ B0a,
                       const float* __restrict__ W0b, const float* __restrict__ B0b,
                       const float* __restrict__ W1a, const float* __restrict__ B1a,
                       const float* __restrict__ W1b, const float* __restrict__ B1b,
                       const float* __restrict__ Wf,  const float* __restrict__ Bf,
                       float* __restrict__ out)
{
    extern __shared__ float smem[];
    float* zre  = &smem[OFF_ZRE];
    float* zim  = &smem[OFF_ZIM];
    float* bt   = &smem[OFF_BT];
    float* mag2s= &smem[OFF_MAG];
    float* wfr  = &smem[OFF_WFR];
    float* wfi  = &smem[OFF_WFI];
    float* w0   = &smem[OFF_W0];
    float* xls  = &smem[OFF_XLS];
    int*   bidx = (int*)&smem[OFF_BIDX];

    const int tid  = threadIdx.x;
    const int c    = blockIdx.y;
    const int m0   = blockIdx.x * TILE_M;
    const int ch   = indices[c];        /* input channel gather */
    const int g    = model_idx[c];      /* per-channel weight group */
    const int wci  = g * NCHAN + c;

    /* ---------------- Phase A: stage small operands ---------------- */
    for (int e = tid; e < 4 * HID; e += 256) {            /* W0a/W0b rows */
        int which = e >> 8, i = (e >> 7) & 1, h = e & 127;
        const float* W0 = which ? W0b : W0a;
        w0[e] = W0[((wci * 2) + i) * HID + h];
    }
    for (int e = tid; e < HID * OUTF; e += 256) {         /* Wf re/im planes */
        const float* p = Wf + ((size_t)wci * HID * OUTF + e) * 2;
        wfr[e] = p[0]; wfi[e] = p[1];
    }
    if (tid < TILE_M) {
        bidx[tid]        = bias_idx[m0 + tid];
        xls[2 * tid + 0] = inp[((size_t)ch * NPTS + m0 + tid) * 2 + 0];
        xls[2 * tid + 1] = inp[((size_t)ch * NPTS + m0 + tid) * 2 + 1];
    }
    __syncthreads();

    /* ---------------- Phase B: layer 0 + first Gabor ---------------- */
    for (int e = tid; e < TILE_M * HID; e += 256) {
        int m = e >> 7, h = e & 127;
        float x0 = xls[2 * m], x1 = xls[2 * m + 1];
        int bi = bidx[m];
        float la = fmaf(x0, w0[h],       fmaf(x1, w0[HID + h],     B0a[bi * HID + h]));
        float lb = fmaf(x0, w0[2*HID+h], fmaf(x1, w0[3 * HID + h], B0b[bi * HID + h]));
        float amp = __expf(-S0SQ * (la * la + lb * lb));
        float s, cc; __sincosf(OMEGA0 * la, &s, &cc);
        zre[m * ZSTR + h] = amp * cc;
        zim[m * ZSTR + h] = amp * s;
    }
    __syncthreads();

    /* ---------------- Phase C: layer 1 complex GEMM via f32 WMMA -----
     * B planes are stored K-pair-interleaved:
     *   plane[(k>>1)*(2*HID) + 2*n + (k&1)] = W[k][n]
     * so a B fragment {W[k][n], W[k+1][n]} is one aligned ds_load_b64. */
    const int lane  = tid & 31;
    const int w     = tid >> 5;
    const int mt    = w & 3;            /* M-subtile 0..3 (16 rows each) */
    const int useB  = w >> 2;           /* 0: lin (W1a), 1: lin2 (W1b)  */
    const int lhalf = lane >> 4;        /* 0/1 */
    const int lmod  = lane & 15;
    const int mrow  = mt * 16 + lmod;   /* A-fragment row */

    const float* W1A = W1a + (size_t)wci * HID * HID * 2;
    const float* W1B = W1b + (size_t)wci * HID * HID * 2;

    v8f cre[8], cim[8];
#pragma unroll
    for (int nt = 0; nt < 8; ++nt) { cre[nt] = (v8f)(0.0f); cim[nt] = (v8f)(0.0f); }

    float2 ra[8], rb[8];                /* global->LDS staging regs */

    auto LOADT = [&](int t) {
#pragma unroll
        for (int i = 0; i < 8; ++i) {
            int e = tid + i * 256;      /* e == k*128 + n within tile */
            size_t go = ((size_t)(t * KT) * HID + e) * 2;
            ra[i] = *(const float2*)&W1A[go];
            rb[i] = *(const float2*)&W1B[go];
        }
    };
    auto STORET = [&](int buf) {
        float* base = &bt[buf * 4 * PLANE];
#pragma unroll
        for (int i = 0; i < 8; ++i) {
            int e  = tid + i * 256;
            int k  = e >> 7, n = e & 127;
            int po = ((k >> 1) * (2 * HID)) + 2 * n + (k & 1);
            base[0 * PLANE + po] = ra[i].x;   /* Wa re */
            base[1 * PLANE + po] = ra[i].y;   /* Wa im */
            base[2 * PLANE + po] = rb[i].x;   /* Wb re */
            base[3 * PLANE + po] = rb[i].y;   /* Wb im */
        }
    };

    LOADT(0);
    STORET(0);
    __syncthreads();

    for (int t = 0; t < NKT; ++t) {
        if (t + 1 < NKT) LOADT(t + 1);
        if (t + 2 < NKT)                 /* CDNA5 global_prefetch_b8 for t+2 */
            __builtin_prefetch(&W1A[((size_t)((t + 2) * KT) * HID + tid * 8) * 2], 0, 1);

        const float* wrp = &bt[(t & 1) * 4 * PLANE + (useB ? 2 : 0) * PLANE];
        const float* wip = wrp + PLANE;

#pragma unroll
        for (int ks = 0; ks < 4; ++ks) {
            int kloc = ks * 4 + 2 * lhalf;          /* K within tile (even) */
            int kk   = t * KT + kloc;               /* absolute K           */
            v2f arf = *(const v2f*)&zre[mrow * ZSTR + kk];
            v2f aif = *(const v2f*)&zim[mrow * ZSTR + kk];
            v2f ain = -aif;                          /* f32 WMMA has no A-neg */
            int prow = (kloc >> 1) * (2 * HID);
#pragma unroll
            for (int nt = 0; nt < 8; ++nt) {
                int n = nt * 16 + lmod;
                v2f brf = *(const v2f*)&wrp[prow + 2 * n];
                v2f bif = *(const v2f*)&wip[prow + 2 * n];
                cre[nt] = __builtin_amdgcn_wmma_f32_16x16x4_f32(
                    false, arf, false, brf, (short)0, cre[nt], false, false);
                cre[nt] = __builtin_amdgcn_wmma_f32_16x16x4_f32(
                    false, ain, false, bif, (short)0, cre[nt], false, false);
                cim[nt] = __builtin_amdgcn_wmma_f32_16x16x4_f32(
                    false, arf, false, bif, (short)0, cim[nt], false, false);
                cim[nt] = __builtin_amdgcn_wmma_f32_16x16x4_f32(
                    false, aif, false, brf, (short)0, cim[nt], false, false);
            }
        }
        if (t + 1 < NKT) STORET((t + 1) & 1);
        __syncthreads();
    }

    /* ---------------- Phase D: bias + second Gabor ------------------- */
    const float* B1 = useB ? B1b : B1a;
#pragma unroll
    for (int nt = 0; nt < 8; ++nt) {
#pragma unroll
        for (int v = 0; v < 8; ++v) {
            int ml = mt * 16 + v + 8 * lhalf;     /* C/D layout: M = v (+8 hi) */
            int n  = nt * 16 + lmod;
            float2 bv = *(const float2*)&B1[((size_t)bidx[ml] * HID + n) * 2];
            cre[nt][v] += bv.x;
            cim[nt][v] += bv.y;
        }
    }
    if (useB) {                                    /* publish |lin2|^2 */
#pragma unroll
        for (int nt = 0; nt < 8; ++nt)
#pragma unroll
            for (int v = 0; v < 8; ++v) {
                int ml = mt * 16 + v + 8 * lhalf;
                int n  = nt * 16 + lmod;
                float r = cre[nt][v], q = cim[nt][v];
                mag2s[ml * HID + n] = r * r + q * q;
            }
    }
    __syncthreads();
    if (!useB) {                                   /* z2 = Gabor(lin, |lin2|^2) */
#pragma unroll
        for (int nt = 0; nt < 8; ++nt)
#pragma unroll
            for (int v = 0; v < 8; ++v) {
                int ml = mt * 16 + v + 8 * lhalf;
                int n  = nt * 16 + lmod;
                float lr = cre[nt][v], li = cim[nt][v];
                float m2 = S0SQ * (lr * lr + li * li + mag2s[ml * HID + n]);
                float amp = __expf(-(OMEGA0 * li) - m2);
                float s, cc; __sincosf(OMEGA0 * lr, &s, &cc);
                zre[ml * ZSTR + n] = amp * cc;
                zim[ml * ZSTR + n] = amp * s;
            }
    }
    __syncthreads();

    /* ---------------- Phase E: final 128->3 projection (real part) --- */
    for (int e = tid; e < TILE_M * OUTF; e += 256) {
        int m = e / OUTF, o = e - m * OUTF;
        float acc = 0.0f;
#pragma unroll 4
        for (int k = 0; k < HID; ++k)
            acc += zre[m * ZSTR + k] * wfr[k * OUTF + o]
                 - zim[m * ZSTR + k] * wfi[k * OUTF + o];
        acc += Bf[((size_t)bidx[m] * OUTF + o) * 2];   /* Re(Bf) */
        out[((size_t)c * NPTS + m0 + m) * OUTF + o] = acc;
    }
}

extern "C" void kernel_launch(void* const* d_in, const int* in_sizes, int n_in,
                              void* d_out, int out_size, void* d_ws, size_t ws_size,
                              hipStream_t stream) {
    (void)in_sizes; (void)n_in; (void)out_size; (void)d_ws; (void)ws_size;
    const float* inp       = (const float*)d_in[0];
    const int*   indices   = (const int*)  d_in[1];
    const int*   model_idx = (const int*)  d_in[2];
    const int*   bias_idx  = (const int*)  d_in[3];
    const float* W0a = (const float*)d_in[4];
    const float* B0a = (const float*)d_in[5];
    const float* W0b = (const float*)d_in[6];
    const float* B0b = (const float*)d_in[7];
    const float* W1a = (const float*)d_in[8];   /* complex64 interleaved */
    const float* B1a = (const float*)d_in[9];
    const float* W1b = (const float*)d_in[10];
    const float* B1b = (const float*)d_in[11];
    const float* Wf  = (const float*)d_in[12];
    const float* Bf  = (const float*)d_in[13];
    float* out = (float*)d_out;

    (void)hipFuncSetAttribute((const void*)wire_fused_kernel,
                              hipFuncAttributeMaxDynamicSharedMemorySize,
                              SMEM_BYTES);
    dim3 grid(NPTS / TILE_M, NCHAN);
    wire_fused_kernel<<<grid, 256, SMEM_BYTES, stream>>>(
        inp, indices, model_idx, bias_idx,
        W0a, B0a, W0b, B0b, W1a, B1a, W1b, B1b, Wf, Bf, out);
}